// EncodeProcessDecode_4947802325262
// MI455X (gfx1250) — compile-verified
//
#include <hip/hip_runtime.h>
#include <math.h>

#define L 128
#define NNODES 50000
#define NEDGES 400000
#define TW 5
#define NSTEPS 8

typedef __attribute__((ext_vector_type(2))) float v2f;
typedef __attribute__((ext_vector_type(8))) float v8f;

__device__ __forceinline__ float reluf(float x) { return x > 0.f ? x : 0.f; }

// ---------------------------------------------------------------------------
// Repack W [K x 128] row-major into WMMA B-fragment order so the GEMM inner
// loop issues ONE coalesced global_load_b64 per k-step:
//   Wp[kstep*256 + ct*32 + lane] = { W[(4*kstep+kb)*128 + n],
//                                    W[(4*kstep+kb+1)*128 + n] }
//   n = ct*16 + (lane&15), kb = (lane>>4)*2
// ---------------------------------------------------------------------------
__global__ void k_repack(const float* __restrict__ W, v2f* __restrict__ Wp,
                         int K) {
  const int idx = blockIdx.x * blockDim.x + threadIdx.x;
  const int kstep = idx >> 8;
  if (kstep < (K >> 2)) {
    const int lane = idx & 31;
    const int ct = (idx >> 5) & 7;
    const int n = ct * 16 + (lane & 15);
    const int kb = (lane >> 4) << 1;
    const int k = kstep << 2;
    v2f b;
    b.x = W[(k + kb) * L + n];
    b.y = W[(k + kb + 1) * L + n];
    Wp[idx] = b;
  }
}

// ---------------------------------------------------------------------------
// WMMA fused layer: H = relu(A @ W + bias), W pre-packed (see k_repack).
//   A: [ROWS x K] in LDS, stride lda (pad +4 -> conflict-free ds_load_b64:
//      16 rows x 4 consecutive dwords tile all 64 banks)
//   H: [ROWS x 128] in LDS, stride ldh
// 256 threads = 8 waves; wave w owns column-tile w. Each B fragment load is
// reused across ROWS/16 row-tiles (v_wmma_f32_16x16x4_f32 accumulate).
// ---------------------------------------------------------------------------
template <int ROWS>
__device__ __forceinline__ void mlp_layer_wmma(
    const float* Alds, int lda, const v2f* __restrict__ Wp,
    const float* __restrict__ bias, float* Hlds, int ldh, int K, int tid) {
  const int wave = tid >> 5;  // column tile
  const int lane = tid & 31;
  const int m = lane & 15;
  const int kb = (lane >> 4) << 1;  // 0 or 2
  constexpr int RT = ROWS / 16;
  v8f acc[RT];
#pragma unroll
  for (int r = 0; r < RT; ++r) acc[r] = (v8f){};
  const v2f* Wpc = Wp + wave * 32 + lane;
  const float* Arow = Alds + m * lda + kb;
  for (int k = 0; k < K; k += 4) {
    const v2f b = Wpc[(k >> 2) * 256];  // coalesced global_load_b64
#pragma unroll
    for (int r = 0; r < RT; ++r) {
      v2f a = *(const v2f*)(Arow + r * 16 * lda + k);  // ds_load_b64
      acc[r] = __builtin_amdgcn_wmma_f32_16x16x4_f32(false, a, false, b,
                                                     (short)0, acc[r], false,
                                                     false);
    }
  }
  const int col = wave * 16 + m;
  const float bb = bias[col];
#pragma unroll
  for (int r = 0; r < RT; ++r) {
    const int rbase = r * 16 + ((lane >> 4) << 3);
#pragma unroll
    for (int v = 0; v < 8; ++v) {
      Hlds[(rbase + v) * ldh + col] = reluf(acc[r][v] + bb);
    }
  }
}

// Wave-parallel LayerNorm stats over a 128-wide LDS row (4 elems/lane).
__device__ __forceinline__ void ln_stats(const float* Hrow, int lane, float& mu,
                                         float& inv) {
  float v0 = Hrow[lane], v1 = Hrow[lane + 32];
  float v2 = Hrow[lane + 64], v3 = Hrow[lane + 96];
  float s = v0 + v1 + v2 + v3;
  float ss = v0 * v0 + v1 * v1 + v2 * v2 + v3 * v3;
#pragma unroll
  for (int off = 16; off; off >>= 1) {
    s += __shfl_xor(s, off, 32);
    ss += __shfl_xor(ss, off, 32);
  }
  mu = s * (1.f / L);
  float var = ss * (1.f / L) - mu * mu;
  inv = rsqrtf(var + 1e-5f);
}

// ---------------------------------------------------------------------------
// Node encoder: [hs,T] (K=2) -> L -> L -> LN. 32 nodes/block (RT=2).
// ---------------------------------------------------------------------------
__global__ void __launch_bounds__(256) k_encode_nodes(
    const float* __restrict__ heat, const float* __restrict__ temp,
    const float* __restrict__ Wn1, const float* __restrict__ bn1,
    const v2f* __restrict__ pWn2, const float* __restrict__ bn2,
    const float* __restrict__ gn, const float* __restrict__ betn,
    float* __restrict__ nodeOut) {
  constexpr int R = 32, LD = L + 4;
  __shared__ float sH1[R * LD];
  __shared__ float sH2[R * LD];
  const int tid = threadIdx.x;
  const int n0 = blockIdx.x * R;
  for (int rr = tid >> 7; rr < R; rr += 2) {
    const int c = tid & (L - 1);
    const int nid = n0 + rr;
    float hs = 0.f, tp = 0.f;
    if (nid < NNODES) { hs = heat[nid]; tp = temp[nid]; }
    sH1[rr * LD + c] = reluf(hs * Wn1[c] + tp * Wn1[L + c] + bn1[c]);
  }
  __syncthreads();
  mlp_layer_wmma<R>(sH1, LD, pWn2, bn2, sH2, LD, L, tid);
  __syncthreads();
  const int wave = tid >> 5, lane = tid & 31;
  for (int r = wave * (R / 8); r < wave * (R / 8) + (R / 8); ++r) {
    const int nid = n0 + r;
    float mu, inv;
    ln_stats(&sH2[r * LD], lane, mu, inv);
    if (nid < NNODES) {
#pragma unroll
      for (int i = 0; i < 4; ++i) {
        const int c = lane + i * 32;
        nodeOut[(size_t)nid * L + c] =
            (sH2[r * LD + c] - mu) * inv * gn[c] + betn[c];
      }
    }
  }
}

// ---------------------------------------------------------------------------
// Edge encoder: [rel(3),|rel|,dT] (K=5) -> L -> L -> LN. 32 edges/block.
// ---------------------------------------------------------------------------
__global__ void __launch_bounds__(256) k_encode_edges(
    const float* __restrict__ mesh, const float* __restrict__ temp,
    const int* __restrict__ snd, const int* __restrict__ rcv,
    const float* __restrict__ We1, const float* __restrict__ be1,
    const v2f* __restrict__ pWe2, const float* __restrict__ be2,
    const float* __restrict__ ge, const float* __restrict__ bete,
    float* __restrict__ edgeOut) {
  constexpr int R = 32, LD = L + 4;
  __shared__ float sF[R * 8];
  __shared__ float sH1[R * LD];
  __shared__ float sH2[R * LD];
  const int tid = threadIdx.x;
  const int e0 = blockIdx.x * R;
  if (tid < R) {
    const int e = e0 + tid;
    const int s = snd[e], r = rcv[e];
    const float rx = mesh[s * 3 + 0] - mesh[r * 3 + 0];
    const float ry = mesh[s * 3 + 1] - mesh[r * 3 + 1];
    const float rz = mesh[s * 3 + 2] - mesh[r * 3 + 2];
    sF[tid * 8 + 0] = rx;
    sF[tid * 8 + 1] = ry;
    sF[tid * 8 + 2] = rz;
    sF[tid * 8 + 3] = sqrtf(rx * rx + ry * ry + rz * rz);
    sF[tid * 8 + 4] = temp[s] - temp[r];
  }
  __syncthreads();
  for (int rr = tid >> 7; rr < R; rr += 2) {
    const int c = tid & (L - 1);
    float x = be1[c];
#pragma unroll
    for (int k = 0; k < 5; ++k) x += sF[rr * 8 + k] * We1[k * L + c];
    sH1[rr * LD + c] = reluf(x);
  }
  __syncthreads();
  mlp_layer_wmma<R>(sH1, LD, pWe2, be2, sH2, LD, L, tid);
  __syncthreads();
  const int wave = tid >> 5, lane = tid & 31;
  for (int r = wave * (R / 8); r < wave * (R / 8) + (R / 8); ++r) {
    float mu, inv;
    ln_stats(&sH2[r * LD], lane, mu, inv);
    const int e = e0 + r;
#pragma unroll
    for (int i = 0; i < 4; ++i) {
      const int c = lane + i * 32;
      edgeOut[(size_t)e * L + c] =
          (sH2[r * LD + c] - mu) * inv * ge[c] + bete[c];
    }
  }
}

// ---------------------------------------------------------------------------
// GraphNetBlock edge update: ein=[n_s,n_r,e] (K=384) -> L -> L -> LN,
// edge += message (in-place: old edge staged in LDS), scatter-add to aggr
// via global_atomic_add_f32. 16 edges/block, b128 staging.
// ---------------------------------------------------------------------------
__global__ void __launch_bounds__(256) k_edge_block(
    const float* __restrict__ node, float* __restrict__ edge,
    const int* __restrict__ snd, const int* __restrict__ rcv,
    const v2f* __restrict__ pW1, const float* __restrict__ b1,
    const v2f* __restrict__ pW2, const float* __restrict__ b2,
    const float* __restrict__ g, const float* __restrict__ bt,
    float* __restrict__ aggr) {
  constexpr int R = 16, K1 = 3 * L, ELD = K1 + 4, LD = L + 4;
  __shared__ float sEin[R * ELD];
  __shared__ float sH1[R * LD];
  __shared__ float sH2[R * LD];
  __shared__ int sRecv[R];
  const int tid = threadIdx.x;
  const int e0 = blockIdx.x * R;
  if (tid < R) sRecv[tid] = rcv[e0 + tid];
  for (int rr = tid >> 5; rr < R; rr += 8) {
    const int c4 = (tid & 31) << 2;
    const int e = e0 + rr;
    const int s = snd[e];
    const int r2 = rcv[e];
    *(float4*)&sEin[rr * ELD + c4] = *(const float4*)&node[(size_t)s * L + c4];
    *(float4*)&sEin[rr * ELD + L + c4] =
        *(const float4*)&node[(size_t)r2 * L + c4];
    *(float4*)&sEin[rr * ELD + 2 * L + c4] =
        *(const float4*)&edge[(size_t)e * L + c4];
  }
  __syncthreads();
  mlp_layer_wmma<R>(sEin, ELD, pW1, b1, sH1, LD, K1, tid);
  __syncthreads();
  mlp_layer_wmma<R>(sH1, LD, pW2, b2, sH2, LD, L, tid);
  __syncthreads();
  const int wave = tid >> 5, lane = tid & 31;
  for (int r = wave * (R / 8); r < wave * (R / 8) + (R / 8); ++r) {
    float mu, inv;
    ln_stats(&sH2[r * LD], lane, mu, inv);
    const int e = e0 + r;
    const int rc = sRecv[r];
#pragma unroll
    for (int i = 0; i < 4; ++i) {
      const int c = lane + i * 32;
      const float ne = (sH2[r * LD + c] - mu) * inv * g[c] + bt[c];
      edge[(size_t)e * L + c] = ne + sEin[r * ELD + 2 * L + c];
      unsafeAtomicAdd(&aggr[(size_t)rc * L + c], ne);
    }
  }
}

// ---------------------------------------------------------------------------
// GraphNetBlock node update: [n,aggr] (K=256) -> L -> L -> LN, + residual.
// In-place (block reads/writes only its own 16 rows). 16 nodes/block.
// ---------------------------------------------------------------------------
__global__ void __launch_bounds__(256) k_node_block(
    float* __restrict__ node, const float* __restrict__ aggr,
    const v2f* __restrict__ pW1, const float* __restrict__ b1,
    const v2f* __restrict__ pW2, const float* __restrict__ b2,
    const float* __restrict__ g, const float* __restrict__ bt) {
  constexpr int R = 16, K1 = 2 * L, ALD = K1 + 4, LD = L + 4;
  __shared__ float sA[R * ALD];
  __shared__ float sH1[R * LD];
  __shared__ float sH2[R * LD];
  const int tid = threadIdx.x;
  const int n0 = blockIdx.x * R;
  for (int rr = tid >> 5; rr < R; rr += 8) {
    const int c4 = (tid & 31) << 2;
    const int nid = n0 + rr;
    *(float4*)&sA[rr * ALD + c4] = *(const float4*)&node[(size_t)nid * L + c4];
    *(float4*)&sA[rr * ALD + L + c4] =
        *(const float4*)&aggr[(size_t)nid * L + c4];
  }
  __syncthreads();
  mlp_layer_wmma<R>(sA, ALD, pW1, b1, sH1, LD, K1, tid);
  __syncthreads();
  mlp_layer_wmma<R>(sH1, LD, pW2, b2, sH2, LD, L, tid);
  __syncthreads();
  const int wave = tid >> 5, lane = tid & 31;
  for (int r = wave * (R / 8); r < wave * (R / 8) + (R / 8); ++r) {
    float mu, inv;
    ln_stats(&sH2[r * LD], lane, mu, inv);
    const int nid = n0 + r;
#pragma unroll
    for (int i = 0; i < 4; ++i) {
      const int c = lane + i * 32;
      node[(size_t)nid * L + c] = (sH2[r * LD + c] - mu) * inv * g[c] + bt[c] +
                                  sA[r * ALD + c];
    }
  }
}

// ---------------------------------------------------------------------------
// Decoder: L->8 swish ->TW, scale by dt, transpose to [TW, N]. 64 nodes/block.
// ---------------------------------------------------------------------------
__global__ void __launch_bounds__(64) k_decode(
    const float* __restrict__ node, const float* __restrict__ Wd1,
    const float* __restrict__ bd1, const float* __restrict__ Wd2,
    const float* __restrict__ bd2, float* __restrict__ out) {
  constexpr int R = 64, LD = L + 4;
  __shared__ float sN[R * LD];
  const int tid = threadIdx.x;
  const int n0 = blockIdx.x * R;
  for (int idx = tid; idx < R * (L / 4); idx += 64) {
    const int r = idx >> 5, c4 = (idx & 31) << 2;
    const int nid = n0 + r;
    float4 v = make_float4(0.f, 0.f, 0.f, 0.f);
    if (nid < NNODES) v = *(const float4*)&node[(size_t)nid * L + c4];
    *(float4*)&sN[r * LD + c4] = v;
  }
  __syncthreads();
  const int nid = n0 + tid;
  if (nid < NNODES) {
    float h[8];
#pragma unroll
    for (int j = 0; j < 8; ++j) h[j] = bd1[j];
    for (int c = 0; c < L; ++c) {
      const float x = sN[tid * LD + c];
#pragma unroll
      for (int j = 0; j < 8; ++j) h[j] += x * Wd1[c * 8 + j];
    }
#pragma unroll
    for (int j = 0; j < 8; ++j) h[j] = h[j] / (1.f + __expf(-h[j]));
#pragma unroll
    for (int t = 0; t < TW; ++t) {
      float d = bd2[t];
#pragma unroll
      for (int j = 0; j < 8; ++j) d += h[j] * Wd2[j * TW + t];
      out[(size_t)t * NNODES + nid] = d * (float)(t + 1);
    }
  }
}

__global__ void k_zero4(float4* __restrict__ p, size_t n4) {
  const size_t i = (size_t)blockIdx.x * blockDim.x + threadIdx.x;
  if (i < n4) p[i] = make_float4(0.f, 0.f, 0.f, 0.f);
}

extern "C" void kernel_launch(void* const* d_in, const int* in_sizes, int n_in,
                              void* d_out, int out_size, void* d_ws,
                              size_t ws_size, hipStream_t stream) {
  const float* heat = (const float*)d_in[0];
  const float* temp = (const float*)d_in[1];
  const float* mesh = (const float*)d_in[2];
  const int* snd = (const int*)d_in[3];
  const int* rcv = (const int*)d_in[4];
  const float* Wn1 = (const float*)d_in[5];
  const float* bn1 = (const float*)d_in[6];
  const float* Wn2 = (const float*)d_in[7];
  const float* bn2 = (const float*)d_in[8];
  const float* gn = (const float*)d_in[9];
  const float* betn = (const float*)d_in[10];
  const float* We1 = (const float*)d_in[11];
  const float* be1 = (const float*)d_in[12];
  const float* We2 = (const float*)d_in[13];
  const float* be2 = (const float*)d_in[14];
  const float* ge = (const float*)d_in[15];
  const float* bete = (const float*)d_in[16];
  const float* BeW1 = (const float*)d_in[17];
  const float* Beb1 = (const float*)d_in[18];
  const float* BeW2 = (const float*)d_in[19];
  const float* Beb2 = (const float*)d_in[20];
  const float* Beg = (const float*)d_in[21];
  const float* Bebt = (const float*)d_in[22];
  const float* BnW1 = (const float*)d_in[23];
  const float* Bnb1 = (const float*)d_in[24];
  const float* BnW2 = (const float*)d_in[25];
  const float* Bnb2 = (const float*)d_in[26];
  const float* Bng = (const float*)d_in[27];
  const float* Bnbt = (const float*)d_in[28];
  const float* Wd1 = (const float*)d_in[29];
  const float* bd1 = (const float*)d_in[30];
  const float* Wd2 = (const float*)d_in[31];
  const float* bd2 = (const float*)d_in[32];
  float* out = (float*)d_out;

  char* ws = (char*)d_ws;
  const size_t nodeBytes = (size_t)NNODES * L * sizeof(float);
  const size_t edgeBytes = (size_t)NEDGES * L * sizeof(float);
  float* nodeBuf = (float*)ws;
  float* edgeBuf = (float*)(ws + nodeBytes);
  float* aggr = (float*)(ws + nodeBytes + edgeBytes);

  // Packed-weight region (v2f fragments), after aggr.
  // PK(K) = 64*K v2f elements.
  v2f* pk = (v2f*)(ws + nodeBytes + edgeBytes + nodeBytes);
  v2f* pWn2 = pk;               // 64*128
  v2f* pWe2 = pWn2 + 64 * L;    // 64*128
  v2f* pStep = pWe2 + 64 * L;   // per step: 64*(384+128+256+128) = 64*896
  const size_t stepPk = (size_t)64 * 896;

  // --- repack all weights into B-fragment order ---
  k_repack<<<(64 * L + 255) / 256, 256, 0, stream>>>(Wn2, pWn2, L);
  k_repack<<<(64 * L + 255) / 256, 256, 0, stream>>>(We2, pWe2, L);
  for (int s = 0; s < NSTEPS; ++s) {
    v2f* pBeW1 = pStep + s * stepPk;
    v2f* pBeW2 = pBeW1 + 64 * 384;
    v2f* pBnW1 = pBeW2 + 64 * 128;
    v2f* pBnW2 = pBnW1 + 64 * 256;
    k_repack<<<(64 * 384 + 255) / 256, 256, 0, stream>>>(
        BeW1 + (size_t)s * 3 * L * L, pBeW1, 3 * L);
    k_repack<<<(64 * 128 + 255) / 256, 256, 0, stream>>>(
        BeW2 + (size_t)s * L * L, pBeW2, L);
    k_repack<<<(64 * 256 + 255) / 256, 256, 0, stream>>>(
        BnW1 + (size_t)s * 2 * L * L, pBnW1, 2 * L);
    k_repack<<<(64 * 128 + 255) / 256, 256, 0, stream>>>(
        BnW2 + (size_t)s * L * L, pBnW2, L);
  }

  // --- encode ---
  k_encode_nodes<<<(NNODES + 31) / 32, 256, 0, stream>>>(
      heat, temp, Wn1, bn1, pWn2, bn2, gn, betn, nodeBuf);
  k_encode_edges<<<NEDGES / 32, 256, 0, stream>>>(
      mesh, temp, snd, rcv, We1, be1, pWe2, be2, ge, bete, edgeBuf);

  // --- process: 8 GraphNetBlocks ---
  for (int s = 0; s < NSTEPS; ++s) {
    v2f* pBeW1 = pStep + s * stepPk;
    v2f* pBeW2 = pBeW1 + 64 * 384;
    v2f* pBnW1 = pBeW2 + 64 * 128;
    v2f* pBnW2 = pBnW1 + 64 * 256;
    k_zero4<<<((size_t)NNODES * L / 4 + 255) / 256, 256, 0, stream>>>(
        (float4*)aggr, (size_t)NNODES * L / 4);
    k_edge_block<<<NEDGES / 16, 256, 0, stream>>>(
        nodeBuf, edgeBuf, snd, rcv, pBeW1, Beb1 + (size_t)s * L, pBeW2,
        Beb2 + (size_t)s * L, Beg + (size_t)s * L, Bebt + (size_t)s * L, aggr);
    k_node_block<<<NNODES / 16, 256, 0, stream>>>(
        nodeBuf, aggr, pBnW1, Bnb1 + (size_t)s * L, pBnW2,
        Bnb2 + (size_t)s * L, Bng + (size_t)s * L, Bnbt + (size_t)s * L);
  }

  // --- decode ---
  k_decode<<<(NNODES + 63) / 64, 64, 0, stream>>>(nodeBuf, Wd1, bd1, Wd2, bd2,
                                                  out);
}